// GptOssTopKRouter_19954418057882
// MI455X (gfx1250) — compile-verified
//
#include <hip/hip_runtime.h>
#include <math.h>

#define HIDDEN 2880
#define NUM_EXPERTS 32
#define T_TOKENS 16384
#define WAVES_PER_BLOCK 8
#define TOKENS_PER_WAVE 32            // two 16-row M-tiles per wave
#define TOKENS_PER_BLOCK (WAVES_PER_BLOCK * TOKENS_PER_WAVE)

typedef __attribute__((ext_vector_type(2))) float v2f;
typedef __attribute__((ext_vector_type(4))) float v4f;
typedef __attribute__((ext_vector_type(8))) float v8f;

__global__ __launch_bounds__(256) void gptoss_topk_router_kernel(
    const float* __restrict__ x,      // [T, H]
    const float* __restrict__ w,      // [E, H]
    const float* __restrict__ bias,   // [E]
    float* __restrict__ scores,       // [T, E]
    int*   __restrict__ topIdx)       // [T, 4]
{
    __shared__ float lds[WAVES_PER_BLOCK * TOKENS_PER_WAVE * NUM_EXPERTS]; // 32 KB

    const int tid  = threadIdx.x;
    const int wave = tid >> 5;
    const int lane = tid & 31;
    const int half = lane >> 4;   // which 16-lane half
    const int l16  = lane & 15;
    const int tokenBase = blockIdx.x * TOKENS_PER_BLOCK + wave * TOKENS_PER_WAVE;

    // A (16x4 f32): lane l holds row M=l%16, K = {2*half, 2*half+1}
    const float* aPtr0 = x + (size_t)(tokenBase + l16) * HIDDEN + half * 2;       // tokens 0..15
    const float* aPtr1 = x + (size_t)(tokenBase + 16 + l16) * HIDDEN + half * 2;  // tokens 16..31
    // B (4x16 f32) = W^T: lane l holds column N=l%16 (expert), same K pairing
    const float* b0Ptr = w + (size_t)l16 * HIDDEN + half * 2;          // experts 0..15
    const float* b1Ptr = w + (size_t)(l16 + 16) * HIDDEN + half * 2;   // experts 16..31

    v8f c00 = {};  // A-tile0 x experts 0..15
    v8f c01 = {};  // A-tile0 x experts 16..31
    v8f c10 = {};  // A-tile1 x experts 0..15
    v8f c11 = {};  // A-tile1 x experts 16..31

    #pragma unroll 4
    for (int k = 0; k < HIDDEN; k += 4) {
        v2f a0 = *(const v2f*)(aPtr0 + k);
        v2f a1 = *(const v2f*)(aPtr1 + k);
        v2f b0 = *(const v2f*)(b0Ptr + k);
        v2f b1 = *(const v2f*)(b1Ptr + k);
        c00 = __builtin_amdgcn_wmma_f32_16x16x4_f32(false, a0, false, b0,
                                                    (short)0, c00, false, false);
        c01 = __builtin_amdgcn_wmma_f32_16x16x4_f32(false, a0, false, b1,
                                                    (short)0, c01, false, false);
        c10 = __builtin_amdgcn_wmma_f32_16x16x4_f32(false, a1, false, b0,
                                                    (short)0, c10, false, false);
        c11 = __builtin_amdgcn_wmma_f32_16x16x4_f32(false, a1, false, b1,
                                                    (short)0, c11, false, false);
    }

    const float bb0 = bias[l16];
    const float bb1 = bias[l16 + 16];

    // C/D layout: VGPR r, lane l -> row = r + 8*half, col = l16 (c*1: +16)
    float* wrow = lds + wave * (TOKENS_PER_WAVE * NUM_EXPERTS);
    #pragma unroll
    for (int r = 0; r < 8; ++r) {
        const int m = r + half * 8;
        wrow[m * NUM_EXPERTS + l16]              = c00[r] + bb0;
        wrow[m * NUM_EXPERTS + 16 + l16]         = c01[r] + bb1;
        wrow[(16 + m) * NUM_EXPERTS + l16]       = c10[r] + bb0;
        wrow[(16 + m) * NUM_EXPERTS + 16 + l16]  = c11[r] + bb1;
    }
    __syncthreads();

    // One token per lane (all 32 lanes active)
    const float* row = wrow + lane * NUM_EXPERTS;

    // Stable top-4 (strict >, earlier index wins ties, descending order)
    float tv0 = -INFINITY, tv1 = -INFINITY, tv2 = -INFINITY, tv3 = -INFINITY;
    int   ti0 = 0, ti1 = 0, ti2 = 0, ti3 = 0;
    #pragma unroll
    for (int e = 0; e < NUM_EXPERTS; ++e) {
        const float v = row[e];
        if (v > tv3) {
            if (v > tv0)      { tv3=tv2; ti3=ti2; tv2=tv1; ti2=ti1; tv1=tv0; ti1=ti0; tv0=v; ti0=e; }
            else if (v > tv1) { tv3=tv2; ti3=ti2; tv2=tv1; ti2=ti1; tv1=v;   ti1=e; }
            else if (v > tv2) { tv3=tv2; ti3=ti2; tv2=v;   ti2=e; }
            else              { tv3=v;   ti3=e; }
        }
    }

    // Softmax over the 4 selected logits
    const float e1 = __expf(tv1 - tv0);
    const float e2 = __expf(tv2 - tv0);
    const float e3 = __expf(tv3 - tv0);
    const float inv = 1.0f / (1.0f + e1 + e2 + e3);
    const float p0 = inv, p1 = e1 * inv, p2 = e2 * inv, p3 = e3 * inv;

    const int t = tokenBase + lane;
    float* orow = scores + (size_t)t * NUM_EXPERTS;
    #pragma unroll
    for (int e = 0; e < NUM_EXPERTS; e += 4) {
        v4f vv = {};
        #pragma unroll
        for (int j = 0; j < 4; ++j) {
            const int ee = e + j;
            float v = 0.0f;
            v = (ee == ti0) ? p0 : v;
            v = (ee == ti1) ? p1 : v;
            v = (ee == ti2) ? p2 : v;
            v = (ee == ti3) ? p3 : v;
            vv[j] = v;
        }
        *(v4f*)(orow + e) = vv;   // 16B-aligned: t*32 floats = 128B boundary
    }
    int4 iv; iv.x = ti0; iv.y = ti1; iv.z = ti2; iv.w = ti3;
    *(int4*)(topIdx + (size_t)t * 4) = iv;
}

extern "C" void kernel_launch(void* const* d_in, const int* in_sizes, int n_in,
                              void* d_out, int out_size, void* d_ws, size_t ws_size,
                              hipStream_t stream) {
    const float* x    = (const float*)d_in[0];  // hidden_states [4,4096,2880] f32
    const float* w    = (const float*)d_in[1];  // weight        [32,2880]    f32
    const float* bias = (const float*)d_in[2];  // bias          [32]         f32

    float* scores = (float*)d_out;                                   // [16384,32] f32
    int*   topIdx = (int*)(scores + (size_t)T_TOKENS * NUM_EXPERTS); // [16384,4]  i32

    dim3 grid(T_TOKENS / TOKENS_PER_BLOCK);  // 64 blocks
    dim3 block(256);                          // 8 wave32
    gptoss_topk_router_kernel<<<grid, block, 0, stream>>>(x, w, bias, scores, topIdx);
}